// whole_network_86191403696164
// MI455X (gfx1250) — compile-verified
//
#include <hip/hip_runtime.h>
#include <hip/hip_bf16.h>

typedef __attribute__((ext_vector_type(2))) float v2f;
typedef __attribute__((ext_vector_type(8))) float v8f;
typedef __attribute__((ext_vector_type(4))) unsigned int u32x4;
typedef __attribute__((ext_vector_type(8))) int i32x8;
typedef __attribute__((ext_vector_type(4))) int i32x4;

#define BATCH 8192

// ---------------------------------------------------------------------------
// f32 WMMA 16x16x4 tile GEMM helper (wave32).
// A layout (16x4 f32): lane<16 -> M=lane, holds K=k,k+1 ; lane>=16 -> K=k+2,k+3
// B layout (4x16 f32): lane<16 -> N=lane, holds K rows k,k+1 ; lane>=16 -> k+2,k+3
// C/D (16x16 f32): VGPR j: lanes0-15 -> (M=j, N=lane), lanes16-31 -> (M=j+8)
// B element (k,n) fetched at Bm[k*ldbK + n*ldbN] so transposed operands are free.
// ---------------------------------------------------------------------------
template <int K>
__device__ __forceinline__ v8f wmma_f32_gemm_tile(const float* A, int lda,
                                                  const float* Bm, int ldbK, int ldbN,
                                                  int m0, int n0, v8f acc) {
  const int lane = threadIdx.x & 31;
  const int half = lane >> 4;
  const int l = lane & 15;
  const float* arow = A + (m0 + l) * lda;
#pragma unroll
  for (int k = 0; k < K; k += 4) {
    const int ka = k + 2 * half;
    v2f a, b;
    a.x = arow[ka];
    a.y = arow[ka + 1];
    b.x = Bm[ka * ldbK + (n0 + l) * ldbN];
    b.y = Bm[(ka + 1) * ldbK + (n0 + l) * ldbN];
    acc = __builtin_amdgcn_wmma_f32_16x16x4_f32(false, a, false, b, (short)0, acc,
                                                false, false);
  }
  return acc;
}

__device__ __forceinline__ void wmma_store_tile(float* C, int ldc, int m0, int n0, v8f acc) {
  const int lane = threadIdx.x & 31;
  const int half = lane >> 4;
  const int l = lane & 15;
#pragma unroll
  for (int j = 0; j < 8; ++j) C[(m0 + j + 8 * half) * ldc + n0 + l] = acc[j];
}

// ---------------------------------------------------------------------------
// Tensor Data Mover: async-load a 2-D tile [rows x rowElems] of f32 from
// global memory into LDS, inserting (padAmtM1+1) DWORDs of padding after
// every row (row length must be (2<<padIntLog) DWORDs), i.e. the LDS tile
// gets leading dimension rowElems + padAmtM1 + 1 — matching the padded
// leading dims used by the WMMA tiles to avoid LDS bank conflicts.
// D# layout per CDNA5 ISA ch.8: group0 = {count, lds_addr, global_addr, type},
// group1 = {flags/pad, tensor dims, tile dims, dim0 stride}, groups2/3 = 0
// (2-D tensor). Issued once per wave; completion tracked with TENSORcnt.
// ---------------------------------------------------------------------------
__device__ __forceinline__ void tdm_load_2d(const void* gsrc, void* ldst,
                                            unsigned rowElems, unsigned rows,
                                            unsigned padIntLog, unsigned padAmtM1) {
  unsigned long long ga = (unsigned long long)(uintptr_t)gsrc;
  unsigned lds = (unsigned)(uintptr_t)ldst;  // LDS aperture: addr[31:0] = LDS offset
  u32x4 g0;
  g0[0] = 1u;                                            // count=1, user descriptor
  g0[1] = lds;                                           // lds_addr
  g0[2] = (unsigned)ga;                                  // global_addr[31:0]
  g0[3] = (unsigned)((ga >> 32) & 0x01FFFFFFull)         // global_addr[56:32]
          | (2u << 30);                                  // type = 2 ("image")
  i32x8 g1;
  g1[0] = (int)((2u << 16)                               // data_size = 4 bytes
                | (1u << 20)                             // pad_enable
                | (padIntLog << 22)                      // pad_interval: 2<<v DWORDs
                | (padAmtM1 << 25));                     // pad_amount: v+1 DWORDs
  g1[1] = (int)((rowElems & 0xFFFFu) << 16);             // tensor_dim0[15:0]
  g1[2] = (int)(((rowElems >> 16) & 0xFFFFu)             // tensor_dim0[31:16]
                | ((rows & 0xFFFFu) << 16));             // tensor_dim1[15:0]
  g1[3] = (int)(((rows >> 16) & 0xFFFFu)                 // tensor_dim1[31:16]
                | ((rowElems & 0xFFFFu) << 16));         // tile_dim0
  g1[4] = (int)(rows & 0xFFFFu);                         // tile_dim1 ; tile_dim2 = 0
  g1[5] = (int)rowElems;                                 // tensor_dim0_stride[31:0]
  g1[6] = 0;
  g1[7] = 0;
  i32x4 gz = {0, 0, 0, 0};
#if defined(__clang_major__) && (__clang_major__ >= 23)
  i32x8 gz8 = {0, 0, 0, 0, 0, 0, 0, 0};
  __builtin_amdgcn_tensor_load_to_lds(g0, g1, gz, gz, gz8, 0);
#else
  __builtin_amdgcn_tensor_load_to_lds(g0, g1, gz, gz, 0);
#endif
}

// ---------------------------------------------------------------------------
// Kernel 1: attention + correlation prep.
// One workgroup (256 thr = 8 waves) per batch element.
// Produces xn[b, 100, 32] = row-centered, L2-normalized attention features so
// that corr = clip(xn @ xn^T) can be rebuilt cheaply in each GCN layer.
// LDS arena (floats), aliased:
//   [0      , 7616 )  xs   [112][68]   (x via TDM, padded rows zeroed)
//   [7616   , 14144)  wlds [96][68]    (Wq|Wk|Wv)
//   [0      , 12992)  S    [112][116]  (aliases xs+wlds after QKV done)
//   [14144  , 18176)  Q    [112][36]
//   [18176  , 22208)  K    [112][36]
//   [22208  , 26240)  V    [112][36]
//   [14144  , 18176)  feat [112][36]   (aliases Q after S done)
// ---------------------------------------------------------------------------
__global__ __launch_bounds__(256) void attn_corr_kernel(
    const float* __restrict__ x,  // [B,100,64]
    const float* __restrict__ Wq, const float* __restrict__ bq,
    const float* __restrict__ Wk, const float* __restrict__ bk,
    const float* __restrict__ Wv, const float* __restrict__ bv,
    float* __restrict__ xn)  // [B,100,32]
{
  __shared__ float smem[26240];
  float* xs = smem;                 // [112][68]
  float* wlds = smem + 7616;        // [96][68]
  float* S = smem;                  // [112][116]
  float* Q = smem + 14144;          // [112][36]
  float* Km = smem + 18176;         // [112][36]
  float* V = smem + 22208;          // [112][36]
  float* feat = smem + 14144;       // [112][36]

  const int b = blockIdx.x;
  const int wave = threadIdx.x >> 5;
  const float* xb = x + (size_t)b * 100 * 64;

  // --- TDM: x[b] (100x64, row-padded to ld=68) into xs; wave 0 issues ---
  if (wave == 0) tdm_load_2d(xb, xs, 64, 100, /*2<<5=64 dwords*/ 5, /*+4 dw*/ 3);

  // --- meanwhile: zero pad rows of xs and stage projection weights ---
  for (int i = threadIdx.x; i < 12 * 68; i += 256)
    xs[(100 + i / 68) * 68 + (i % 68)] = 0.f;
  for (int i = threadIdx.x; i < 96 * 64; i += 256) {
    int r = i >> 6, c = i & 63;
    const float* Wsrc = (r < 32) ? Wq : (r < 64) ? Wk : Wv;
    wlds[r * 68 + c] = Wsrc[(r & 31) * 64 + c];
  }
  if (wave == 0) __builtin_amdgcn_s_wait_tensorcnt(0);
  __syncthreads();

  // --- Q/K/V = x @ W^T + b : 3 mats x 7 Mtiles x 2 Ntiles, K=64 ---
  for (int t = wave; t < 42; t += 8) {
    int mat = t / 14;
    int rem = t % 14;
    int mt = rem >> 1, nt = rem & 1;
    v8f acc = {};
    // B(k,n) = W[n][k] -> wlds[(mat*32+n)*68 + k]
    acc = wmma_f32_gemm_tile<64>(xs, 68, wlds + mat * 32 * 68, 1, 68, mt * 16, nt * 16, acc);
    float* dst = (mat == 0) ? Q : (mat == 1) ? Km : V;
    const float* bias = (mat == 0) ? bq : (mat == 1) ? bk : bv;
    const int lane = threadIdx.x & 31, half = lane >> 4, l = lane & 15;
    float bvv = bias[nt * 16 + l];
#pragma unroll
    for (int j = 0; j < 8; ++j)
      dst[(mt * 16 + j + 8 * half) * 36 + nt * 16 + l] = acc[j] + bvv;
  }
  __syncthreads();

  // --- S = Q @ K^T : 7x7 tiles, K=32 ---
  for (int t = wave; t < 49; t += 8) {
    int mt = t / 7, nt = t % 7;
    v8f acc = {};
    acc = wmma_f32_gemm_tile<32>(Q, 36, Km, 1, 36, mt * 16, nt * 16, acc);
    wmma_store_tile(S, 116, mt * 16, nt * 16, acc);
  }
  __syncthreads();

  // --- softmax over columns 0..99 (scale 1/32); zero pad columns ---
  for (int r = threadIdx.x; r < 100; r += 256) {
    float* row = S + r * 116;
    float mx = -1e30f;
    for (int c = 0; c < 100; ++c) mx = fmaxf(mx, row[c]);
    mx *= (1.f / 32.f);
    float sum = 0.f;
    for (int c = 0; c < 100; ++c) {
      float e = __expf(row[c] * (1.f / 32.f) - mx);
      row[c] = e;
      sum += e;
    }
    float inv = 1.f / sum;
    for (int c = 0; c < 100; ++c) row[c] *= inv;
    for (int c = 100; c < 112; ++c) row[c] = 0.f;
  }
  __syncthreads();

  // --- feat = S @ V : 7x2 tiles, K=112 ---
  for (int t = wave; t < 14; t += 8) {
    int mt = t >> 1, nt = t & 1;
    v8f acc = {};
    acc = wmma_f32_gemm_tile<112>(S, 116, V, 36, 1, mt * 16, nt * 16, acc);
    wmma_store_tile(feat, 36, mt * 16, nt * 16, acc);
  }
  __syncthreads();

  // --- xn = (feat - rowmean) / ||feat - rowmean|| ---
  float* xnb = xn + (size_t)b * 100 * 32;
  for (int r = threadIdx.x; r < 100; r += 256) {
    const float* f = feat + r * 36;
    float mean = 0.f;
    for (int d = 0; d < 32; ++d) mean += f[d];
    mean *= (1.f / 32.f);
    float ss = 0.f;
    for (int d = 0; d < 32; ++d) {
      float xc = f[d] - mean;
      ss += xc * xc;
    }
    float inv = rsqrtf(ss);
    for (int d = 0; d < 32; ++d) xnb[r * 32 + d] = (f[d] - mean) * inv;
  }
}

// ---------------------------------------------------------------------------
// Kernel 2: one GCN layer.  z = (clip(xn xn^T) @ h) @ W^T + b
// h = x for the first layer, else tanh((z_prev - mean)*invstd) using the
// batch-wide stats produced by bn_stats_kernel.
// Also emits per-(batch,node) partial sums of z and z^2 for the next BN.
// ---------------------------------------------------------------------------
template <int FIN, int FOUT, bool FIRST>
__global__ __launch_bounds__(256) void gcn_kernel(
    const float* __restrict__ xn,        // [B,100,32]
    const float* __restrict__ hin,       // FIRST: x [B,100,64]; else z_prev [B,100,FIN]
    const float* __restrict__ stats_in,  // [100][2] (mean, invstd) of z_prev
    const float* __restrict__ W,         // [FOUT][FIN]
    const float* __restrict__ bias,      // [FOUT]
    float* __restrict__ zout,            // [B,100,FOUT]
    float* __restrict__ psum,            // [B,100]
    float* __restrict__ psq)             // [B,100]
{
  constexpr int LDH = (FIN > 32) ? 68 : 36;
  constexpr int NT_G = (FIN + 15) / 16;   // N tiles of g (cols = FIN)
  constexpr int NT_Z = (FOUT + 15) / 16;  // N tiles of z (cols = FOUT)
  constexpr int NPW = NT_Z * 16;

  __shared__ float xs[112 * 36];     // xn staging, reused as z staging
  __shared__ float corr[112 * 116];
  __shared__ float hbuf[112 * LDH];
  __shared__ float gbuf[112 * LDH];
  __shared__ float wl[NPW * 68];

  const int b = blockIdx.x;
  const int wave = threadIdx.x >> 5;

  // --- TDM: xn[b] (100x32, row-padded to ld=36) into xs; x for layer 1 ---
  const float* xnb = xn + (size_t)b * 100 * 32;
  if (wave == 0) {
    tdm_load_2d(xnb, xs, 32, 100, /*2<<4=32 dwords*/ 4, /*+4 dw*/ 3);
    if (FIRST)
      tdm_load_2d(hin + (size_t)b * 100 * 64, hbuf, 64, 100, 5, 3);
  }
  // --- zero pad rows of xs ---
  for (int i = threadIdx.x; i < 12 * 36; i += 256)
    xs[(100 + i / 36) * 36 + (i % 36)] = 0.f;
  // --- stage W (rows >= FOUT zeroed) ---
  for (int i = threadIdx.x; i < NPW * FIN; i += 256) {
    int n = i / FIN, k = i % FIN;
    wl[n * 68 + k] = (n < FOUT) ? W[n * FIN + k] : 0.f;
  }
  // --- stage h ---
  if (FIRST) {
    for (int i = threadIdx.x; i < 12 * 68; i += 256)
      hbuf[(100 + i / 68) * 68 + (i % 68)] = 0.f;  // zero pad rows (body via TDM)
  } else {
    const float* zb = hin + (size_t)b * 100 * FIN;
    for (int i = threadIdx.x; i < 112 * FIN; i += 256) {
      int r = i / FIN, c = i % FIN;
      float v = 0.f;
      if (r < 100) {
        float mean = stats_in[2 * r], invs = stats_in[2 * r + 1];
        v = tanhf((zb[r * FIN + c] - mean) * invs);
      }
      hbuf[r * LDH + c] = v;
    }
  }
  if (wave == 0) __builtin_amdgcn_s_wait_tensorcnt(0);
  __syncthreads();

  // --- corr = clip(xn @ xn^T, -1, 1) : 7x7 tiles, K=32 ---
  for (int t = wave; t < 49; t += 8) {
    int mt = t / 7, nt = t % 7;
    v8f acc = {};
    acc = wmma_f32_gemm_tile<32>(xs, 36, xs, 1, 36, mt * 16, nt * 16, acc);
    const int lane = threadIdx.x & 31, half = lane >> 4, l = lane & 15;
#pragma unroll
    for (int j = 0; j < 8; ++j) {
      float v = fminf(1.f, fmaxf(-1.f, acc[j]));
      corr[(mt * 16 + j + 8 * half) * 116 + nt * 16 + l] = v;
    }
  }
  __syncthreads();

  // --- g = corr @ h : 7 x NT_G tiles, K=112 ---
  for (int t = wave; t < 7 * NT_G; t += 8) {
    int mt = t / NT_G, nt = t % NT_G;
    v8f acc = {};
    acc = wmma_f32_gemm_tile<112>(corr, 116, hbuf, LDH, 1, mt * 16, nt * 16, acc);
    wmma_store_tile(gbuf, LDH, mt * 16, nt * 16, acc);
  }
  __syncthreads();

  // --- z = g @ W^T + b : 7 x NT_Z tiles, K=FIN ; stage z into xs ---
  for (int t = wave; t < 7 * NT_Z; t += 8) {
    int mt = t / NT_Z, nt = t % NT_Z;
    v8f acc = {};
    acc = wmma_f32_gemm_tile<FIN>(gbuf, LDH, wl, 1, 68, mt * 16, nt * 16, acc);
    const int lane = threadIdx.x & 31, half = lane >> 4, l = lane & 15;
    const int col = nt * 16 + l;
    const float bvv = (col < FOUT) ? bias[col] : 0.f;
#pragma unroll
    for (int j = 0; j < 8; ++j)
      xs[(mt * 16 + j + 8 * half) * 36 + col] = acc[j] + bvv;
  }
  __syncthreads();

  // --- write z + per-node partial sums for batch-norm ---
  float* zb = zout + (size_t)b * 100 * FOUT;
  for (int i = threadIdx.x; i < 100 * FOUT; i += 256)
    zb[i] = xs[(i / FOUT) * 36 + (i % FOUT)];
  for (int r = threadIdx.x; r < 100; r += 256) {
    float s = 0.f, ss = 0.f;
    for (int c = 0; c < FOUT; ++c) {
      float v = xs[r * 36 + c];
      s += v;
      ss += v * v;
    }
    psum[(size_t)b * 100 + r] = s;
    psq[(size_t)b * 100 + r] = ss;
  }
}

// ---------------------------------------------------------------------------
// Kernel 3: reduce partial sums -> per-node (mean, invstd).  100 blocks.
// ---------------------------------------------------------------------------
__global__ __launch_bounds__(256) void bn_stats_kernel(
    const float* __restrict__ psum, const float* __restrict__ psq,
    float* __restrict__ stats, float count_inv) {
  __shared__ float s1[256];
  __shared__ float s2[256];
  const int n = blockIdx.x;
  float a = 0.f, c = 0.f;
  for (int b = threadIdx.x; b < BATCH; b += 256) {
    a += psum[(size_t)b * 100 + n];
    c += psq[(size_t)b * 100 + n];
  }
  s1[threadIdx.x] = a;
  s2[threadIdx.x] = c;
  __syncthreads();
  for (int off = 128; off > 0; off >>= 1) {
    if (threadIdx.x < off) {
      s1[threadIdx.x] += s1[threadIdx.x + off];
      s2[threadIdx.x] += s2[threadIdx.x + off];
    }
    __syncthreads();
  }
  if (threadIdx.x == 0) {
    float mean = s1[0] * count_inv;
    float var = s2[0] * count_inv - mean * mean;
    stats[2 * n] = mean;
    stats[2 * n + 1] = rsqrtf(var + 1e-5f);
  }
}

// ---------------------------------------------------------------------------
// Kernel 4: classifier head.  out[b,:] = tanh(bn(z5[b])).flatten() @ Wf^T + bf
// ---------------------------------------------------------------------------
__global__ __launch_bounds__(256) void head_kernel(
    const float* __restrict__ z5,     // [B,100,2]
    const float* __restrict__ stats,  // [100][2]
    const float* __restrict__ Wf,     // [2][200]
    const float* __restrict__ bf,     // [2]
    float* __restrict__ out)          // [B,2]
{
  __shared__ float wf_s[400];
  __shared__ float st_s[200];
  for (int i = threadIdx.x; i < 400; i += 256) wf_s[i] = Wf[i];
  for (int i = threadIdx.x; i < 200; i += 256) st_s[i] = stats[i];
  __syncthreads();

  const int b = blockIdx.x * 256 + threadIdx.x;
  if (b >= BATCH) return;
  const float* zb = z5 + (size_t)b * 200;
  float a0 = bf[0], a1 = bf[1];
  for (int n = 0; n < 100; ++n) {
    float mean = st_s[2 * n], invs = st_s[2 * n + 1];
    float h0 = tanhf((zb[2 * n] - mean) * invs);
    float h1 = tanhf((zb[2 * n + 1] - mean) * invs);
    a0 += h0 * wf_s[2 * n] + h1 * wf_s[2 * n + 1];
    a1 += h0 * wf_s[200 + 2 * n] + h1 * wf_s[200 + 2 * n + 1];
  }
  out[(size_t)b * 2 + 0] = a0;
  out[(size_t)b * 2 + 1] = a1;
}

// ---------------------------------------------------------------------------
extern "C" void kernel_launch(void* const* d_in, const int* in_sizes, int n_in,
                              void* d_out, int out_size, void* d_ws, size_t ws_size,
                              hipStream_t stream) {
  (void)in_sizes; (void)n_in; (void)out_size; (void)ws_size;
  const float* x  = (const float*)d_in[0];
  const float* Wq = (const float*)d_in[1];
  const float* bq = (const float*)d_in[2];
  const float* Wk = (const float*)d_in[3];
  const float* bk = (const float*)d_in[4];
  const float* Wv = (const float*)d_in[5];
  const float* bv = (const float*)d_in[6];
  const float* W1 = (const float*)d_in[7];
  const float* b1 = (const float*)d_in[8];
  const float* W2 = (const float*)d_in[9];
  const float* b2 = (const float*)d_in[10];
  const float* W3 = (const float*)d_in[11];
  const float* b3 = (const float*)d_in[12];
  const float* W4 = (const float*)d_in[13];
  const float* b4 = (const float*)d_in[14];
  const float* W5 = (const float*)d_in[15];
  const float* b5 = (const float*)d_in[16];
  const float* Wf = (const float*)d_in[17];
  const float* bf = (const float*)d_in[18];

  char* p = (char*)d_ws;
  float* xn = (float*)p; p += (size_t)BATCH * 100 * 32 * sizeof(float);
  float* zA = (float*)p; p += (size_t)BATCH * 100 * 32 * sizeof(float);
  float* zB = (float*)p; p += (size_t)BATCH * 100 * 16 * sizeof(float);
  float* ps = (float*)p; p += (size_t)BATCH * 100 * sizeof(float);
  float* pq = (float*)p; p += (size_t)BATCH * 100 * sizeof(float);
  float* st = (float*)p;  // 5 * 200 floats

  dim3 blk(256);
  attn_corr_kernel<<<BATCH, blk, 0, stream>>>(x, Wq, bq, Wk, bk, Wv, bv, xn);

  gcn_kernel<64, 32, true ><<<BATCH, blk, 0, stream>>>(xn, x,  nullptr,  W1, b1, zA, ps, pq);
  bn_stats_kernel<<<100, blk, 0, stream>>>(ps, pq, st + 0,   1.f / (BATCH * 32.f));
  gcn_kernel<32, 16, false><<<BATCH, blk, 0, stream>>>(xn, zA, st + 0,   W2, b2, zB, ps, pq);
  bn_stats_kernel<<<100, blk, 0, stream>>>(ps, pq, st + 200, 1.f / (BATCH * 16.f));
  gcn_kernel<16,  8, false><<<BATCH, blk, 0, stream>>>(xn, zB, st + 200, W3, b3, zA, ps, pq);
  bn_stats_kernel<<<100, blk, 0, stream>>>(ps, pq, st + 400, 1.f / (BATCH * 8.f));
  gcn_kernel< 8,  4, false><<<BATCH, blk, 0, stream>>>(xn, zA, st + 400, W4, b4, zB, ps, pq);
  bn_stats_kernel<<<100, blk, 0, stream>>>(ps, pq, st + 600, 1.f / (BATCH * 4.f));
  gcn_kernel< 4,  2, false><<<BATCH, blk, 0, stream>>>(xn, zB, st + 600, W5, b5, zA, ps, pq);
  bn_stats_kernel<<<100, blk, 0, stream>>>(ps, pq, st + 800, 1.f / (BATCH * 2.f));

  head_kernel<<<(BATCH + 255) / 256, blk, 0, stream>>>(zA, st + 800, Wf, bf, (float*)d_out);
}